// GravNetLayer_36421322670786
// MI455X (gfx1250) — compile-verified
//
#include <hip/hip_runtime.h>
#include <hip/hip_bf16.h>
#include <math.h>

typedef __attribute__((ext_vector_type(2))) float v2f;
typedef __attribute__((ext_vector_type(8))) float v8f;

// ---------------------------------------------------------------------------
// WMMA helper: D = A(16x4, f32) * B(4x16, f32) + C(16x16, f32)
// A layout: lane l<16 holds row M=l, {K=0,K=1}; lane l>=16 holds row M=l-16, {K=2,K=3}
// B layout: lane l<16 holds col N=l, {K=0,K=1}; lane l>=16 holds col N=l-16, {K=2,K=3}
// C/D layout: vgpr v, lane l<16 -> (M=v, N=l); lane l>=16 -> (M=v+8, N=l-16)
// ---------------------------------------------------------------------------
__device__ __forceinline__ v8f wmma_f32_4(v2f a, v2f b, v8f c) {
    return __builtin_amdgcn_wmma_f32_16x16x4_f32(
        /*neg_a=*/false, a, /*neg_b=*/false, b,
        /*c_mod=*/(short)0, c, /*reuse_a=*/false, /*reuse_b=*/false);
}

// ---------------------------------------------------------------------------
// Kernel 1: x_feats = X[M,K] @ W[K,N] + bias   (M=16384, K=64, N=128)
// One wave computes one 16x16 output tile. 8 waves (256 thr) per block.
// ---------------------------------------------------------------------------
__global__ void feat_gemm_kernel(const float* __restrict__ X,
                                 const float* __restrict__ W,
                                 const float* __restrict__ bias,
                                 float* __restrict__ out,
                                 int M, int K, int N) {
    const int lane = threadIdx.x & 31;
    const int waveInBlk = threadIdx.x >> 5;
    const int wavesPerBlk = blockDim.x >> 5;
    const int waveId = blockIdx.x * wavesPerBlk + waveInBlk;
    const int tilesN = N >> 4;
    const int tileM = waveId / tilesN;
    const int tileN = waveId % tilesN;
    if (tileM * 16 >= M) return;

    const int l = lane & 15;
    const int half = lane >> 4;          // 0: K pair {k,k+1}; 1: {k+2,k+3}
    const int rowA = tileM * 16 + l;
    const int colB = tileN * 16 + l;

    v8f acc = {};
    for (int k = 0; k < K; k += 4) {
        const int ka = k + half * 2;
        v2f a = *(const v2f*)(X + (size_t)rowA * K + ka);   // 8B aligned (ka even)
        v2f b;
        b.x = W[(size_t)ka * N + colB];
        b.y = W[(size_t)(ka + 1) * N + colB];
        acc = wmma_f32_4(a, b, acc);
    }

    const float bv = bias[colB];
#pragma unroll
    for (int v = 0; v < 8; ++v) {
        const int row = tileM * 16 + v + half * 8;
        out[(size_t)row * N + colB] = acc[v] + bv;
    }
}

// ---------------------------------------------------------------------------
// Kernel 2: coords = X @ W_space + b_space (S=4), squared norms, and the
// mask-folded candidate norm sqm = mask ? ||c||^2 : +inf  (so that
// d = sr + sqm[m] - 2<c_r,c_m> reproduces the reference's masked distance).
// ---------------------------------------------------------------------------
__global__ void coords_kernel(const float* __restrict__ X,
                              const float* __restrict__ Ws,
                              const float* __restrict__ bs,
                              const unsigned char* __restrict__ mask,
                              float* __restrict__ coords,
                              float* __restrict__ sq,
                              float* __restrict__ sqm,
                              int M, int K) {
    const int row = blockIdx.x * blockDim.x + threadIdx.x;
    if (row >= M) return;
    float c0 = bs[0], c1 = bs[1], c2 = bs[2], c3 = bs[3];
    const float* xr = X + (size_t)row * K;
    for (int k = 0; k < K; ++k) {
        const float xv = xr[k];
        const float4 w = *(const float4*)(Ws + 4 * k);
        c0 = fmaf(xv, w.x, c0);
        c1 = fmaf(xv, w.y, c1);
        c2 = fmaf(xv, w.z, c2);
        c3 = fmaf(xv, w.w, c3);
    }
    float4 c = make_float4(c0, c1, c2, c3);
    *(float4*)(coords + (size_t)row * 4) = c;
    const float s = c0 * c0 + c1 * c1 + c2 * c2 + c3 * c3;
    sq[row] = s;
    sqm[row] = mask[row] ? s : INFINITY;
}

// ---------------------------------------------------------------------------
// Kernel 3: per-node top-16 nearest neighbors (includes self, dist 0).
// All 256 rows of a block scan the same per-batch candidate set, so
// candidates are staged through LDS in 256-wide tiles (float4 coords + norm);
// inner-loop LDS reads are uniform across the wave -> conflict-free broadcast.
// Register-resident sorted top-16, static indexing only (no scratch).
// ---------------------------------------------------------------------------
#define KNN_TILE 256

__global__ void knn_kernel(const float* __restrict__ coords,
                           const float* __restrict__ sq,
                           const float* __restrict__ sqm,
                           int* __restrict__ knn,
                           int N) {
    __shared__ float4 cbuf[KNN_TILE];
    __shared__ float  sbuf[KNN_TILE];

    const int tid = threadIdx.x;                       // 0..255
    const int row = blockIdx.x * blockDim.x + tid;
    const int base = (row / N) * N;                    // uniform per block (256 | 4096)

    const float4 cr = *(const float4*)(coords + (size_t)row * 4);
    const float sr = sq[row];

    float bd[16];
    int   bi[16];
#pragma unroll
    for (int j = 0; j < 16; ++j) { bd[j] = INFINITY; bi[j] = base; }

    for (int t0 = 0; t0 < N; t0 += KNN_TILE) {
        const int c = base + t0 + tid;
        cbuf[tid] = *(const float4*)(coords + (size_t)c * 4);
        sbuf[tid] = sqm[c];
        __syncthreads();

        for (int m = 0; m < KNN_TILE; ++m) {
            const float4 cm = cbuf[m];                 // LDS broadcast
            const float dot = cr.x * cm.x + cr.y * cm.y + cr.z * cm.z + cr.w * cm.w;
            const float d = sr + sbuf[m] - 2.0f * dot; // masked -> +inf automatically
            if (d < bd[15]) {
                const int idx = base + t0 + m;
                // sorted insert, all indices compile-time constant after unroll
#pragma unroll
                for (int j = 15; j > 0; --j) {
                    if (d < bd[j - 1])      { bd[j] = bd[j - 1]; bi[j] = bi[j - 1]; }
                    else if (d < bd[j])     { bd[j] = d;         bi[j] = idx;       }
                }
                if (d < bd[0])              { bd[0] = d;         bi[0] = idx;       }
            }
        }
        __syncthreads();
    }

    int* ko = knn + (size_t)row * 16;
#pragma unroll
    for (int j = 0; j < 16; ++j) ko[j] = bi[j];
}

// ---------------------------------------------------------------------------
// Kernel 4: y = x_feats + mean_{j in knn}(x_feats[j])  (masked nodes: agg=0)
// float4-vectorized: 8 rows per 256-thread block, 32 threads x float4 per row.
// ---------------------------------------------------------------------------
__global__ void agg_kernel(const float* __restrict__ xfeats,
                           const int* __restrict__ knn,
                           const unsigned char* __restrict__ mask,
                           float* __restrict__ y,
                           int F) {
    const int tid = threadIdx.x;
    const int row = blockIdx.x * 8 + (tid >> 5);
    const int col = (tid & 31) * 4;
    const int* ki = knn + (size_t)row * 16;

    float4 s = make_float4(0.f, 0.f, 0.f, 0.f);
#pragma unroll
    for (int j = 0; j < 16; ++j) {
        const float4 v = *(const float4*)(xfeats + (size_t)ki[j] * F + col);
        s.x += v.x; s.y += v.y; s.z += v.z; s.w += v.w;
    }
    const float sc = mask[row] ? (1.0f / 16.0f) : 0.0f;
    const float4 x0 = *(const float4*)(xfeats + (size_t)row * F + col);
    float4 o = make_float4(fmaf(s.x, sc, x0.x), fmaf(s.y, sc, x0.y),
                           fmaf(s.z, sc, x0.z), fmaf(s.w, sc, x0.w));
    *(float4*)(y + (size_t)row * F + col) = o;
}

// ---------------------------------------------------------------------------
// Kernel 5: out = relu(y@W1 + b1) @ W2 + b2, fused through LDS.
// Block = 256 threads = 8 waves handles 16 rows x 128 cols.
// Wave w computes h-tile (cols 16w..16w+15) via WMMA, writes relu'd h to LDS,
// barrier, then second WMMA chain reads A-operand from LDS (ds_load_b64).
// W2 is prefetched (global_prefetch_b8) while GEMM-1 executes.
// ---------------------------------------------------------------------------
#define MLP_F 128
#define MLP_HS (MLP_F + 4)   // padded LDS row stride (floats) to dodge bank conflicts

__global__ void mlp_fused_kernel(const float* __restrict__ Y,
                                 const float* __restrict__ W1,
                                 const float* __restrict__ b1,
                                 const float* __restrict__ W2,
                                 const float* __restrict__ b2,
                                 float* __restrict__ out,
                                 int M) {
    __shared__ float hbuf[16 * MLP_HS];

    const int lane = threadIdx.x & 31;
    const int wave = threadIdx.x >> 5;   // N-tile id, 0..7
    const int l = lane & 15;
    const int half = lane >> 4;
    const int rowBase = blockIdx.x * 16;
    const int rowA = rowBase + l;
    const int colB = wave * 16 + l;

    // warm W2 (64KB, L2-resident) toward the WGP while GEMM-1 runs
    {
        const float* p = W2 + (size_t)threadIdx.x * 64;   // 256 thr x 256B = full W2
        __builtin_prefetch(p, 0, 1);
        __builtin_prefetch(p + 32, 0, 1);
    }

    // ---- h = relu(y @ W1 + b1) -> LDS ----
    v8f acc = {};
    for (int k = 0; k < MLP_F; k += 4) {
        const int ka = k + half * 2;
        v2f a = *(const v2f*)(Y + (size_t)rowA * MLP_F + ka);
        v2f b;
        b.x = W1[(size_t)ka * MLP_F + colB];
        b.y = W1[(size_t)(ka + 1) * MLP_F + colB];
        acc = wmma_f32_4(a, b, acc);
    }
    const float bv1 = b1[colB];
#pragma unroll
    for (int v = 0; v < 8; ++v) {
        const int r = v + half * 8;
        const float h = acc[v] + bv1;
        hbuf[r * MLP_HS + colB] = h > 0.0f ? h : 0.0f;
    }
    __syncthreads();

    // ---- out = h @ W2 + b2 ----
    v8f acc2 = {};
    for (int k = 0; k < MLP_F; k += 4) {
        const int ka = k + half * 2;
        v2f a = *(const v2f*)(&hbuf[l * MLP_HS + ka]);   // ds_load_b64 (stride & ka even)
        v2f b;
        b.x = W2[(size_t)ka * MLP_F + colB];
        b.y = W2[(size_t)(ka + 1) * MLP_F + colB];
        acc2 = wmma_f32_4(a, b, acc2);
    }
    const float bv2 = b2[colB];
#pragma unroll
    for (int v = 0; v < 8; ++v) {
        const int r = v + half * 8;
        out[(size_t)(rowBase + r) * MLP_F + colB] = acc2[v] + bv2;
    }
}

// ---------------------------------------------------------------------------
// Host-side orchestration
// ---------------------------------------------------------------------------
extern "C" void kernel_launch(void* const* d_in, const int* in_sizes, int n_in,
                              void* d_out, int out_size, void* d_ws, size_t ws_size,
                              hipStream_t stream) {
    (void)in_sizes; (void)n_in; (void)out_size; (void)ws_size;

    const int B = 4, N = 4096, Fin = 64, Fout = 128;
    const int BN = B * N;   // 16384

    const float*         x      = (const float*)d_in[0];
    const unsigned char* mask   = (const unsigned char*)d_in[1];
    const float*         Wspace = (const float*)d_in[2];
    const float*         bspace = (const float*)d_in[3];
    const float*         Wfeat  = (const float*)d_in[4];
    const float*         bfeat  = (const float*)d_in[5];
    const float*         W1     = (const float*)d_in[6];
    const float*         b1     = (const float*)d_in[7];
    const float*         W2     = (const float*)d_in[8];
    const float*         b2     = (const float*)d_in[9];
    float*               outp   = (float*)d_out;

    // workspace layout (bytes, all offsets 256B-aligned)
    char* ws = (char*)d_ws;
    float* xfeats = (float*)(ws);                            //  8 MB: 16384*128 f32
    float* yb     = (float*)(ws + (size_t)8  * 1024 * 1024); //  8 MB
    float* coords = (float*)(ws + (size_t)16 * 1024 * 1024); // 256 KB: 16384*4 f32
    float* sq     = (float*)(ws + (size_t)16 * 1024 * 1024 + 256 * 1024); // 64 KB
    float* sqm    = (float*)(ws + (size_t)16 * 1024 * 1024 + 320 * 1024); // 64 KB
    int*   knn    = (int*)  (ws + (size_t)16 * 1024 * 1024 + 384 * 1024); //  1 MB

    // 1) x_feats = x @ W_feat + b_feat    (WMMA)
    {
        const int tilesM = BN / 16, tilesN = Fout / 16;       // 1024 x 8
        const int waves = tilesM * tilesN;                    // 8192
        const int blocks = waves / 8;                         // 8 waves/block
        feat_gemm_kernel<<<blocks, 256, 0, stream>>>(x, Wfeat, bfeat, xfeats, BN, Fin, Fout);
    }

    // 2) coords + squared norms (+ mask-folded candidate norms)
    coords_kernel<<<(BN + 255) / 256, 256, 0, stream>>>(x, Wspace, bspace, mask,
                                                        coords, sq, sqm, BN, Fin);

    // 3) top-16 nearest neighbors per node (LDS-tiled candidate scan)
    knn_kernel<<<BN / 256, 256, 0, stream>>>(coords, sq, sqm, knn, N);

    // 4) y = x_feats + mean(neighbor feats)   (float4 gathers)
    agg_kernel<<<BN / 8, 256, 0, stream>>>(xfeats, knn, mask, yb, Fout);

    // 5) out = relu(y@W1+b1)@W2 + b2      (fused WMMA through LDS)
    mlp_fused_kernel<<<BN / 16, 256, 0, stream>>>(yb, W1, b1, W2, b2, outp, BN);
}